// simpleLRU_49787260895674
// MI455X (gfx1250) — compile-verified
//
#include <hip/hip_runtime.h>
#include <hip/hip_bf16.h>

// cumsum over axis=1 of (64, 32768, 16) f32.
// 3-phase decoupled scan; phase 3 uses V_WMMA_F32_16X16X4_F32 to compute
// per-tile prefix sums as D = L(16x16 lower-tri ones) * X(16x16) + carry.

typedef __attribute__((ext_vector_type(2))) float v2f;
typedef __attribute__((ext_vector_type(4))) float v4f;
typedef __attribute__((ext_vector_type(8))) float v8f;

#define NBATCH   64
#define LSEQ     32768
#define NCH      16
#define SEG      512                   // timesteps per wave-owned segment
#define NSEG     (LSEQ / SEG)          // 64 segments per batch
#define TOTSEG   (NBATCH * NSEG)       // 4096 segments total
#define SEGFLTS  (SEG * NCH)           // 8192 floats per segment

// ---------------------------------------------------------------------------
// Pass 1: per-segment channel totals. One wave per segment.
// Lane l: channel group g = l&3 (4 channels), row phase r = l>>2 (0..7).
// Each B128 load instruction pulls 512 contiguous bytes across the wave.
// ---------------------------------------------------------------------------
__global__ void seg_sums_kernel(const float* __restrict__ in,
                                float* __restrict__ sums) {
  const int wave = (blockIdx.x * blockDim.x + threadIdx.x) >> 5;
  const int lane = threadIdx.x & 31;
  const int g = lane & 3;
  const int r = lane >> 2;

  const v4f* p = (const v4f*)(in + (size_t)wave * SEGFLTS) + (r * 4 + g);
  v4f acc = {0.f, 0.f, 0.f, 0.f};
#pragma unroll 4
  for (int i = 0; i < SEG / 8; ++i) {
    acc += p[32 * i];  // rows r, r+8, r+16, ... for this channel group
  }
  // Reduce across the 8 lanes (lane bits 2..4) sharing channel group g.
#pragma unroll
  for (int m = 4; m <= 16; m <<= 1) {
    acc[0] += __shfl_xor(acc[0], m, 32);
    acc[1] += __shfl_xor(acc[1], m, 32);
    acc[2] += __shfl_xor(acc[2], m, 32);
    acc[3] += __shfl_xor(acc[3], m, 32);
  }
  if (r == 0) {
    ((v4f*)sums)[wave * 4 + g] = acc;
  }
}

// ---------------------------------------------------------------------------
// Pass 2: per-batch exclusive scan of segment totals (tiny).
// One wave per batch; lanes 0..15 each own a channel, serial over 64 segs.
// ---------------------------------------------------------------------------
__global__ void seg_scan_kernel(const float* __restrict__ sums,
                                float* __restrict__ excl) {
  const int batch = blockIdx.x;
  const int lane = threadIdx.x;
  if (lane < NCH) {
    float carry = 0.f;
    for (int s = 0; s < NSEG; ++s) {
      const int idx = (batch * NSEG + s) * NCH + lane;
      const float v = sums[idx];
      excl[idx] = carry;
      carry += v;
    }
  }
}

// ---------------------------------------------------------------------------
// Pass 3: WMMA tile scan. One wave per segment, 32 tiles of 16 timesteps.
// Per tile: D = L * X + broadcast(carry); carry' = D[15][*].
//   A (16x4 f32, 2 VGPRs): lane half selects K={0,1} / {2,3}; L[m][k]=(k<=m).
//   B (4x16  f32, 2 VGPRs): VGPR v, half h -> row (v + 2h); cols = lane&15.
//   C/D (16x16 f32, 8 VGPRs): VGPR i -> rows i (lanes 0-15), i+8 (lanes 16-31).
// Output stores are non-temporal so writes don't evict the L2-resident input.
// ---------------------------------------------------------------------------
__global__ void wmma_scan_kernel(const float* __restrict__ in,
                                 const float* __restrict__ excl,
                                 float* __restrict__ out) {
  const int wave = (blockIdx.x * blockDim.x + threadIdx.x) >> 5;
  const int lane = threadIdx.x & 31;
  const int m    = lane & 15;   // row (A) / column (B,C,D) index
  const int half = lane >> 4;   // 0: lanes 0-15, 1: lanes 16-31
  const int k0   = half * 2;    // K index held in VGPR0 of A and B

  const float* base  = in  + (size_t)wave * SEGFLTS;
  float*       obase = out + (size_t)wave * SEGFLTS;

  float carry = excl[wave * NCH + m];

  for (int tile = 0; tile < SEG / 16; ++tile) {
    const float* tb = base + tile * 256;

    v8f d;
#pragma unroll
    for (int i = 0; i < 8; ++i) d[i] = carry;  // C = carry broadcast per column

#pragma unroll
    for (int kb = 0; kb < 16; kb += 4) {
      v2f a, b;
      a[0] = ((kb + k0)     <= m) ? 1.0f : 0.0f;   // L[m][kb+k0]
      a[1] = ((kb + k0 + 1) <= m) ? 1.0f : 0.0f;   // L[m][kb+k0+1]
      b[0] = tb[(kb + k0)     * NCH + m];          // X[kb+k0][m]
      b[1] = tb[(kb + k0 + 1) * NCH + m];          // X[kb+k0+1][m]
      d = __builtin_amdgcn_wmma_f32_16x16x4_f32(
          /*neg_a=*/false, a, /*neg_b=*/false, b,
          /*c_mod=*/(short)0, d, /*reuse_a=*/false, /*reuse_b=*/false);
    }

    float* ob = obase + tile * 256;
#pragma unroll
    for (int i = 0; i < 8; ++i) {
      __builtin_nontemporal_store(d[i], &ob[(i + 8 * half) * NCH + m]);
    }

    // carry' = D[15][c] : VGPR 7, lanes 16-31, col = lane-16
    carry = __shfl(d[7], 16 + m, 32);
  }
}

// ---------------------------------------------------------------------------
extern "C" void kernel_launch(void* const* d_in, const int* in_sizes, int n_in,
                              void* d_out, int out_size, void* d_ws, size_t ws_size,
                              hipStream_t stream) {
  const float* in = (const float*)d_in[0];
  float* out = (float*)d_out;

  // Workspace: sums[4096*16] ++ excl[4096*16]  (512 KiB total)
  float* sums = (float*)d_ws;
  float* excl = sums + (size_t)TOTSEG * NCH;

  // 4096 waves -> 512 blocks of 256 threads (8 waves) for the heavy passes.
  seg_sums_kernel<<<TOTSEG / 8, 256, 0, stream>>>(in, sums);
  seg_scan_kernel<<<NBATCH, 32, 0, stream>>>(sums, excl);
  wmma_scan_kernel<<<TOTSEG / 8, 256, 0, stream>>>(in, excl, out);
}